// TextQNetPolicy_22058952032403
// MI455X (gfx1250) — compile-verified
//
#include <hip/hip_runtime.h>

#define B_ 64
#define A_ 4096
#define D_ 512
#define TOPK 5

typedef __attribute__((ext_vector_type(2))) float v2f;
typedef __attribute__((ext_vector_type(8))) float v8f;

// ---------------------------------------------------------------------------
// Kernel 1: logits_raw[b,a] = <s_embed[b,:], a_embeds[b,a,:]>
// One wave32 per 16-action tile, V_WMMA_F32_16X16X4_F32 over K=512 in steps
// of 4.  B-fragment = s-vector chunk broadcast to all 16 columns, so every
// column of D holds the same 16 dot products; we read column N=0 (lanes 0/16).
// Memory-bound: streams a_embeds (512 MB) exactly once.
// ---------------------------------------------------------------------------
__global__ __launch_bounds__(256) void logits_wmma_kernel(
    const float* __restrict__ s_embed,
    const float* __restrict__ a_embeds,
    float* __restrict__ logits_raw)
{
    const int wave  = threadIdx.x >> 5;
    const int lane  = threadIdx.x & 31;
    const int b     = blockIdx.x >> 5;      // 32 blocks per batch row
    const int chunk = blockIdx.x & 31;      // 128 actions per block
    const int tile  = chunk * 8 + wave;     // 16-action tile, 0..255
    const int row   = lane & 15;            // M index within tile
    const int khalf = (lane >> 4) << 1;     // lanes 0-15 -> K+0,1 ; 16-31 -> K+2,3

    const float* __restrict__ arow =
        a_embeds + ((size_t)b * A_ + (size_t)tile * 16 + row) * D_ + khalf;
    const float* __restrict__ svec = s_embed + (size_t)b * D_ + khalf;

    v8f acc = {};
#pragma unroll 8
    for (int k = 0; k < D_; k += 4) {
        if ((k & 31) == 0)
            __builtin_prefetch(arow + k + 64, 0, 1);   // global_prefetch_b8
        v2f afrag = *(const v2f*)(arow + k);           // A: 16x4 tile fragment
        v2f bfrag = *(const v2f*)(svec + k);           // B: s chunk, all cols equal
        acc = __builtin_amdgcn_wmma_f32_16x16x4_f32(
            /*neg_a=*/false, afrag, /*neg_b=*/false, bfrag,
            /*c_mod=*/(short)0, acc, /*reuse_a=*/false, /*reuse_b=*/false);
    }

    // D column N==0 lives in lanes 0 (M=r) and 16 (M=8+r), VGPR r = acc[r].
    if (row == 0) {
        const int mbase = tile * 16 + ((lane >> 4) << 3);
        float* out = logits_raw + (size_t)b * A_ + mbase;
#pragma unroll
        for (int r = 0; r < 8; ++r) out[r] = acc[r];
    }
}

// ---------------------------------------------------------------------------
// Kernel 2: per-row min of raw logits (reference: fill = global min - 1)
// ---------------------------------------------------------------------------
__global__ __launch_bounds__(256) void rowmin_kernel(
    const float* __restrict__ lr, float* __restrict__ rowmin)
{
    __shared__ float sm[256];
    const int b = blockIdx.x, t = threadIdx.x;
    float m = 3.402823466e38f;
    for (int i = t; i < A_; i += 256) m = fminf(m, lr[(size_t)b * A_ + i]);
    sm[t] = m; __syncthreads();
    for (int s = 128; s > 0; s >>= 1) {
        if (t < s) sm[t] = fminf(sm[t], sm[t + s]);
        __syncthreads();
    }
    if (t == 0) rowmin[b] = sm[0];
}

// ---------------------------------------------------------------------------
// Kernel 3: 64 row-mins -> fill value (global min - 1)
// ---------------------------------------------------------------------------
__global__ __launch_bounds__(64) void fill_kernel(
    const float* __restrict__ rowmin, float* __restrict__ fill)
{
    __shared__ float sm[64];
    const int t = threadIdx.x;
    sm[t] = rowmin[t]; __syncthreads();
    for (int s = 32; s > 0; s >>= 1) {
        if (t < s) sm[t] = fminf(sm[t], sm[t + s]);
        __syncthreads();
    }
    if (t == 0) fill[0] = sm[0] - 1.0f;
}

// ---------------------------------------------------------------------------
// Kernel 4: mask+fill -> output logits; iterative top-5 (lowest-index ties);
// temperature softmax over the <=5 kept entries; categorical draw.
// ---------------------------------------------------------------------------
__global__ __launch_bounds__(256) void policy_kernel(
    const float* __restrict__ lr, const int* __restrict__ avail,
    const float* __restrict__ fillp, const int* __restrict__ alpha_p,
    int* __restrict__ action, float* __restrict__ logits_out)
{
    __shared__ float sv[256];
    __shared__ int   si[256];
    __shared__ float selv[TOPK];
    __shared__ int   seli[TOPK];

    const int b = blockIdx.x, t = threadIdx.x;
    const float fill = fillp[0];
    const int base = t * 16;

    float vals[16];
#pragma unroll
    for (int i = 0; i < 16; ++i) {
        const int idx = base + i;
        float v = lr[(size_t)b * A_ + idx];
        v = avail[(size_t)b * A_ + idx] ? v : fill;
        vals[i] = v;
        logits_out[(size_t)b * A_ + idx] = v;   // reference returns masked logits
    }

    for (int it = 0; it < TOPK; ++it) {
        float bv = -3.402823466e38f;
        int   bi = 0x7FFFFFFF;
#pragma unroll
        for (int i = 0; i < 16; ++i) {
            const int idx = base + i;
            bool taken = false;
            for (int j = 0; j < it; ++j) taken |= (seli[j] == idx);
            const float v = vals[i];
            if (!taken && (v > bv || (v == bv && idx < bi))) { bv = v; bi = idx; }
        }
        sv[t] = bv; si[t] = bi; __syncthreads();
        for (int s = 128; s > 0; s >>= 1) {
            if (t < s) {
                const float v2 = sv[t + s]; const int i2 = si[t + s];
                if (v2 > sv[t] || (v2 == sv[t] && i2 < si[t])) { sv[t] = v2; si[t] = i2; }
            }
            __syncthreads();
        }
        if (t == 0) { selv[it] = sv[0]; seli[it] = si[0]; }
        __syncthreads();
    }

    if (t == 0) {
        const float alpha = (float)alpha_p[0];
        const float inv_a = 1.0f / (alpha != 0.0f ? alpha : 1.0f);
        const float maxv  = selv[0];              // row max = top-1 value
        float p[TOPK]; float sum = 0.0f;
        for (int j = 0; j < TOPK; ++j) {
            const int idx = seli[j];
            const bool keep = avail[(size_t)b * A_ + idx] != 0;
            const float e = keep ? __expf((selv[j] - maxv) * inv_a) : 0.0f;
            p[j] = e; sum += e;
        }
        // deterministic per-row uniform (stand-in for jax.random.key(42))
        unsigned h = 42u * 0x9E3779B9u + (unsigned)b;
        h ^= h >> 16; h *= 0x85EBCA6Bu; h ^= h >> 13; h *= 0xC2B2AE35u; h ^= h >> 16;
        const float u = (float)(h >> 8) * (1.0f / 16777216.0f) * sum;
        int a = seli[0]; float c = 0.0f;
        for (int j = 0; j < TOPK; ++j) {
            c += p[j];
            if (p[j] > 0.0f && u <= c) { a = seli[j]; break; }
        }
        action[b] = a;
    }
}

// ---------------------------------------------------------------------------
extern "C" void kernel_launch(void* const* d_in, const int* in_sizes, int n_in,
                              void* d_out, int out_size, void* d_ws, size_t ws_size,
                              hipStream_t stream) {
    const float* s_embed  = (const float*)d_in[0];   // [B, D]
    const float* a_embeds = (const float*)d_in[1];   // [B, A, D]
    const int*   avail    = (const int*)d_in[2];     // [B, A]
    const int*   alpha_p  = (const int*)d_in[3];     // scalar

    float* ws      = (float*)d_ws;
    float* raw     = ws;                             // [B, A] raw logits
    float* rowmin  = ws + (size_t)B_ * A_;           // [B]
    float* fillv   = rowmin + B_;                    // [1]

    int*   action     = (int*)d_out;                 // [B] int32 bits
    float* logits_out = (float*)d_out + B_;          // [B, A]

    logits_wmma_kernel<<<dim3(B_ * 32), dim3(256), 0, stream>>>(s_embed, a_embeds, raw);
    rowmin_kernel<<<dim3(B_), dim3(256), 0, stream>>>(raw, rowmin);
    fill_kernel<<<dim3(1), dim3(64), 0, stream>>>(rowmin, fillv);
    policy_kernel<<<dim3(B_), dim3(256), 0, stream>>>(raw, avail, fillv, alpha_p,
                                                      action, logits_out);
}